// GCN_50302656971357
// MI455X (gfx1250) — compile-verified
//
#include <hip/hip_runtime.h>
#include <hip/hip_bf16.h>

typedef float v2f __attribute__((ext_vector_type(2)));
typedef float v8f __attribute__((ext_vector_type(8)));

#define HID 64

// ---------------------------------------------------------------- utilities
__global__ void zero_f32(float* __restrict__ p, int n) {
  int t = blockIdx.x * blockDim.x + threadIdx.x;
  if (t < n) p[t] = 0.0f;
}

__global__ void degree_kernel(const int* __restrict__ src, const int* __restrict__ dst,
                              float* __restrict__ ns, float* __restrict__ nd, int E) {
  int e = blockIdx.x * blockDim.x + threadIdx.x;
  if (e < E) {
    atomicAdd(&ns[src[e]], 1.0f);
    atomicAdd(&nd[dst[e]], 1.0f);
  }
}

// deg -> deg>0 ? rsqrt(deg) : 0   (in place, over ns and nd back-to-back)
__global__ void rsqrt_kernel(float* __restrict__ p, int n) {
  int t = blockIdx.x * blockDim.x + threadIdx.x;
  if (t < n) {
    float d = p[t];
    p[t] = (d > 0.0f) ? rsqrtf(d) : 0.0f;
  }
}

// ------------------------------------------------- fp32 WMMA tiled GEMM
// Out[m, n] = rowscale[m] * sum_k A[m,k] * W[k,n]  (+ bias[n])
// One wave computes one 16x16 tile via V_WMMA_F32_16X16X4_F32 (K stepped by 4).
// Grid is exact (M % 16 == 0, Ncols % 16 == 0), EXEC all-ones around WMMA.
__global__ void gemm_f32_wmma(const float* __restrict__ A,
                              const float* __restrict__ rowscale,   // may be null
                              const float* __restrict__ W,
                              const float* __restrict__ bias,       // may be null
                              float* __restrict__ Out,
                              int M, int K, int Ncols) {
  const int wave    = (blockIdx.x * blockDim.x + threadIdx.x) >> 5;
  const int lane    = threadIdx.x & 31;
  const int tiles_n = Ncols >> 4;
  const int tm      = wave / tiles_n;
  const int tn      = wave - tm * tiles_n;
  const int half    = lane >> 4;     // 0: lanes 0-15, 1: lanes 16-31
  const int r       = lane & 15;

  const int arow = tm * 16 + r;      // A row this lane feeds
  const int bcol = tn * 16 + r;      // B column this lane feeds
  const float s  = rowscale ? rowscale[arow] : 1.0f;
  const float* __restrict__ Arow = A + (size_t)arow * (size_t)K;

  v8f c = {};
  for (int k = 0; k < K; k += 4) {
    const int kk = k + half * 2;     // per-ISA A/B 16x4 / 4x16 lane layout
    v2f a, b;
    a.x = Arow[kk + 0] * s;
    a.y = Arow[kk + 1] * s;
    b.x = W[(size_t)(kk + 0) * Ncols + bcol];
    b.y = W[(size_t)(kk + 1) * Ncols + bcol];
    c = __builtin_amdgcn_wmma_f32_16x16x4_f32(
        /*neg_a=*/false, a, /*neg_b=*/false, b,
        /*c_mod=*/(short)0, c, /*reuse_a=*/false, /*reuse_b=*/false);
  }

  const float bv   = bias ? bias[bcol] : 0.0f;
  const int  orow0 = tm * 16 + half * 8;   // C/D layout: VGPR i -> row i (+8 for hi half)
#pragma unroll
  for (int i = 0; i < 8; ++i) {
    Out[(size_t)(orow0 + i) * Ncols + bcol] = c[i] + bv;
  }
}

// ----------------------------------------------- edge gather + scatter-add
// 16 consecutive threads handle one edge (one float4 chunk each): the gather of
// h[src] is a fully coalesced 256B row read (L2-resident: 20.5MB << 192MB L2),
// the scatter is 4x global_atomic_add_f32 into agg[dst].
__global__ void scatter_edges(const float* __restrict__ h,
                              const int* __restrict__ src,
                              const int* __restrict__ dst,
                              float* __restrict__ agg, int E) {
  int t = blockIdx.x * blockDim.x + threadIdx.x;
  if (t >= E * 16) return;
  const int e     = t >> 4;
  const int chunk = t & 15;
  const int sIdx  = src[e];
  const int dIdx  = dst[e];
  const float4 v = ((const float4*)(h + (size_t)sIdx * HID))[chunk];
  float* __restrict__ out = agg + (size_t)dIdx * HID + chunk * 4;
  atomicAdd(out + 0, v.x);
  atomicAdd(out + 1, v.y);
  atomicAdd(out + 2, v.z);
  atomicAdd(out + 3, v.w);
}

// h_out = relu(agg * nd[row] + bias[col])
__global__ void finalize_kernel(const float* __restrict__ agg,
                                const float* __restrict__ nd,
                                const float* __restrict__ bias,
                                float* __restrict__ out, int total) {
  int t = blockIdx.x * blockDim.x + threadIdx.x;
  if (t < total) {
    const int row = t >> 6;      // HID == 64
    const int col = t & 63;
    float v = agg[t] * nd[row] + bias[col];
    out[t] = v > 0.0f ? v : 0.0f;
  }
}

// ---------------------------------------------------------------- driver
extern "C" void kernel_launch(void* const* d_in, const int* in_sizes, int n_in,
                              void* d_out, int out_size, void* d_ws, size_t ws_size,
                              hipStream_t stream) {
  const float* x     = (const float*)d_in[0];
  const int*   src   = (const int*)  d_in[1];
  const int*   dst   = (const int*)  d_in[2];
  const float* W0    = (const float*)d_in[3];
  const float* b0    = (const float*)d_in[4];
  const float* W1    = (const float*)d_in[5];
  const float* b1    = (const float*)d_in[6];
  const float* W2    = (const float*)d_in[7];
  const float* b2    = (const float*)d_in[8];
  const float* Wout  = (const float*)d_in[9];
  const float* bout  = (const float*)d_in[10];
  float*       outp  = (float*)d_out;

  const int NFEAT = 128, NCLASS = 32;
  const int N = in_sizes[0] / NFEAT;   // 80000
  const int E = in_sizes[1];           // 1280000

  // workspace layout (16B-aligned slices)
  float* ns = (float*)d_ws;            // [N]
  float* nd = ns + N;                  // [N]
  float* hA = nd + N;                  // [N*HID]
  float* hB = hA + (size_t)N * HID;    // [N*HID]

  const int TB = 256;
  const int nhid_total = N * HID;

  // --- degree normalizers (recomputed each call: deterministic) ---
  zero_f32<<<(2 * N + TB - 1) / TB, TB, 0, stream>>>(ns, 2 * N);
  degree_kernel<<<(E + TB - 1) / TB, TB, 0, stream>>>(src, dst, ns, nd, E);
  rsqrt_kernel<<<(2 * N + TB - 1) / TB, TB, 0, stream>>>(ns, 2 * N);

  // gemm launch geometry: one wave (32 threads) per 16x16 tile, 4 waves/block
  auto gemm_blocks = [&](int M, int Ncols) { return (M / 16) * (Ncols / 16) / 4; };

  // ---------------- layer 0: x(128) -> hA(64) ----------------
  gemm_f32_wmma<<<gemm_blocks(N, HID), 128, 0, stream>>>(x, ns, W0, nullptr, hA, N, NFEAT, HID);
  zero_f32<<<(nhid_total + TB - 1) / TB, TB, 0, stream>>>(hB, nhid_total);
  scatter_edges<<<(E * 16 + TB - 1) / TB, TB, 0, stream>>>(hA, src, dst, hB, E);
  finalize_kernel<<<(nhid_total + TB - 1) / TB, TB, 0, stream>>>(hB, nd, b0, hA, nhid_total);

  // ---------------- layer 1: hA -> hB ----------------
  gemm_f32_wmma<<<gemm_blocks(N, HID), 128, 0, stream>>>(hA, ns, W1, nullptr, hB, N, HID, HID);
  zero_f32<<<(nhid_total + TB - 1) / TB, TB, 0, stream>>>(hA, nhid_total);
  scatter_edges<<<(E * 16 + TB - 1) / TB, TB, 0, stream>>>(hB, src, dst, hA, E);
  finalize_kernel<<<(nhid_total + TB - 1) / TB, TB, 0, stream>>>(hA, nd, b1, hB, nhid_total);

  // ---------------- layer 2: hB -> hA ----------------
  gemm_f32_wmma<<<gemm_blocks(N, HID), 128, 0, stream>>>(hB, ns, W2, nullptr, hA, N, HID, HID);
  zero_f32<<<(nhid_total + TB - 1) / TB, TB, 0, stream>>>(hB, nhid_total);
  scatter_edges<<<(E * 16 + TB - 1) / TB, TB, 0, stream>>>(hA, src, dst, hB, E);
  finalize_kernel<<<(nhid_total + TB - 1) / TB, TB, 0, stream>>>(hB, nd, b2, hA, nhid_total);

  // ---------------- output layer: hA(64) -> out(32), +bias, no relu ----------------
  gemm_f32_wmma<<<gemm_blocks(N, NCLASS), 128, 0, stream>>>(hA, nullptr, Wout, bout, outp,
                                                            N, HID, NCLASS);
}